// PAGCN_50053548867899
// MI455X (gfx1250) — compile-verified
//
#include <hip/hip_runtime.h>
#include <hip/hip_bf16.h>
#include <math.h>

typedef __attribute__((ext_vector_type(2))) float v2f;
typedef __attribute__((ext_vector_type(8))) float v8f;

#define N_USERS 100000
#define N_ITEMS 200000
#define NTOT    300000
#define DIM     64
#define PPERS   4
#define E_SUB   1000000
#define E_SOC   1600000
#define BATCH   4096
#define K3D     192

static const size_t ND = (size_t)NTOT * DIM;      // 19,200,000
static const size_t UD = (size_t)N_USERS * DIM;   // 6,400,000

// ---------------- init: pers[p]=e0, mean=4*e0, usoc=user_emb ----------------
__global__ void k_init(const float* __restrict__ ue, const float* __restrict__ ie,
                       float* __restrict__ pers, float* __restrict__ mean,
                       float* __restrict__ usoc) {
    size_t i = (size_t)blockIdx.x * blockDim.x + threadIdx.x;
    size_t stride = (size_t)gridDim.x * blockDim.x;
    for (; i < ND; i += stride) {
        float e = (i < UD) ? ue[i] : ie[i - UD];
        mean[i] = 4.0f * e;
        pers[i] = e; pers[ND + i] = e; pers[2*ND + i] = e; pers[3*ND + i] = e;
        if (i < UD) usoc[i] = e;
    }
}

__global__ void k_zero(float* __restrict__ p, size_t n) {
    size_t i = (size_t)blockIdx.x * blockDim.x + threadIdx.x;
    size_t stride = (size_t)gridDim.x * blockDim.x;
    for (; i < n; i += stride) p[i] = 0.0f;
}

// ---------------- SpMM: out[dst] += val * x[src], one wave per edge ----------------
__global__ void k_spmm(const int* __restrict__ src, const int* __restrict__ dst,
                       const float* __restrict__ val,
                       const float* __restrict__ xin, float* __restrict__ xout,
                       int nEdges, int edgeStride, size_t nodeStride) {
    int p = blockIdx.y;
    const int*   s = src + (size_t)p * edgeStride;
    const int*   d = dst + (size_t)p * edgeStride;
    const float* v = val + (size_t)p * edgeStride;
    const float* x = xin  + (size_t)p * nodeStride;
    float*       o = xout + (size_t)p * nodeStride;

    int lane   = threadIdx.x & 31;
    int wave   = (int)((blockIdx.x * blockDim.x + threadIdx.x) >> 5);
    int nWaves = (int)((gridDim.x * blockDim.x) >> 5);

    for (int e = wave; e < nEdges; e += nWaves) {
        int   si = s[e];
        int   di = d[e];
        float w  = v[e];
        // prefetch next edge's source row (global_prefetch_b8)
        int en = e + nWaves;
        if (en < nEdges) {
            int sn = s[en];
            __builtin_prefetch(&x[(size_t)sn * DIM + lane], 0, 1);
        }
        float x0 = x[(size_t)si * DIM + lane];
        float x1 = x[(size_t)si * DIM + lane + 32];
        atomicAdd(&o[(size_t)di * DIM + lane],      w * x0);
        atomicAdd(&o[(size_t)di * DIM + lane + 32], w * x1);
    }
}

// -------- sum personas; users -> build concat [u | us | u*us]; items -> mean += --------
__global__ void k_sum_build(const float* __restrict__ pers, const float* __restrict__ usoc,
                            float* __restrict__ mean, float* __restrict__ cbuf,
                            float* __restrict__ sumsq) {
    size_t i = (size_t)blockIdx.x * blockDim.x + threadIdx.x;
    size_t stride = (size_t)gridDim.x * blockDim.x;
    if (i == 0) *sumsq = 0.0f;
    for (; i < ND; i += stride) {
        float s = pers[i] + pers[ND + i] + pers[2*ND + i] + pers[3*ND + i];
        size_t row = i >> 6;
        int    col = (int)(i & 63);
        if (row < N_USERS) {
            float us = usoc[i];
            cbuf[row * K3D + col]        = s;
            cbuf[row * K3D + 64 + col]   = us;
            cbuf[row * K3D + 128 + col]  = s * us;
        } else {
            mean[i] += s;
        }
    }
}

// ---------------- WMMA GEMM: out = act(A[MxK] @ W[KxN] + bias), fp32 exact ----------------
// One wave computes a 16 x (NT*16) strip of columns [colOff, colOff+NT*16) of an
// ldN-wide output. Ping-pong software pipeline (K % 8 == 0): fragment set B for step
// k+4 is loaded while set A's NT WMMAs execute, and vice versa — no copy chains, two
// live fragment sets, accumulators stay in registers (NT<=6 -> ~90 VGPRs, no spills).
template <int NT>
__global__ __launch_bounds__(256, 1)
void k_gemm_wmma(const float* __restrict__ A, const float* __restrict__ W,
                 const float* __restrict__ bias, float* __restrict__ out,
                 int M, int K, int ldN, int colOff, int doTanh, float* sumsq) {
    int wave   = (int)((blockIdx.x * blockDim.x + threadIdx.x) >> 5);
    int lane   = threadIdx.x & 31;
    int mTiles = M >> 4;
    if (wave >= mTiles) return;             // wave-uniform exit (EXEC all-ones for WMMA)

    int half = lane >> 4;                   // 0: lanes 0-15, 1: lanes 16-31
    int l16  = lane & 15;
    const float* Arow = A + (size_t)(wave * 16 + l16) * K;
    const float* Wc   = W + colOff + l16;

    v8f acc[NT];
#pragma unroll
    for (int n = 0; n < NT; ++n) acc[n] = (v8f){};

    v2f aA, aB, bA[NT], bB[NT];
    // prologue: set A <- k = 0
    {
        int ka = 2 * half;                  // lanes 0-15: K=k,k+1 ; lanes 16-31: K=k+2,k+3
        aA.x = Arow[ka];
        aA.y = Arow[ka + 1];
#pragma unroll
        for (int n = 0; n < NT; ++n) {
            bA[n].x = Wc[(size_t)ka * ldN + n * 16];
            bA[n].y = Wc[(size_t)(ka + 1) * ldN + n * 16];
        }
    }

    for (int k = 0; k < K; k += 8) {
        // set B <- k+4 (always in range: k+4 <= K-4), issued before set A's WMMAs
        {
            int ka = k + 4 + 2 * half;
            aB.x = Arow[ka];
            aB.y = Arow[ka + 1];
#pragma unroll
            for (int n = 0; n < NT; ++n) {
                bB[n].x = Wc[(size_t)ka * ldN + n * 16];
                bB[n].y = Wc[(size_t)(ka + 1) * ldN + n * 16];
            }
        }
#pragma unroll
        for (int n = 0; n < NT; ++n)
            acc[n] = __builtin_amdgcn_wmma_f32_16x16x4_f32(
                false, aA, false, bA[n], (short)0, acc[n], false, false);

        // set A <- k+8 (skip on last iteration), issued before set B's WMMAs
        if (k + 8 < K) {                    // wave-uniform
            int ka = k + 8 + 2 * half;
            aA.x = Arow[ka];
            aA.y = Arow[ka + 1];
#pragma unroll
            for (int n = 0; n < NT; ++n) {
                bA[n].x = Wc[(size_t)ka * ldN + n * 16];
                bA[n].y = Wc[(size_t)(ka + 1) * ldN + n * 16];
            }
        }
#pragma unroll
        for (int n = 0; n < NT; ++n)
            acc[n] = __builtin_amdgcn_wmma_f32_16x16x4_f32(
                false, aB, false, bB[n], (short)0, acc[n], false, false);
    }

    float local = 0.0f;
#pragma unroll
    for (int n = 0; n < NT; ++n) {
        int   col = colOff + n * 16 + l16;
        float bb  = bias[col];
#pragma unroll
        for (int r = 0; r < 8; ++r) {
            int   row = wave * 16 + r + 8 * half;  // VGPR r: lanes0-15->M=r, lanes16-31->M=r+8
            float vv  = acc[n][r] + bb;
            if (doTanh) vv = tanhf(vv);
            out[(size_t)row * ldN + col] = vv;
            local += vv * vv;
        }
    }
    if (sumsq) {
        for (int off = 16; off > 0; off >>= 1)
            local += __shfl_down(local, off, 32);
        if (lane == 0) atomicAdd(sumsq, local);
    }
}

// ---------------- mean[user part] += t3 * rsqrt(sumsq) ----------------
__global__ void k_norm_accum(const float* __restrict__ t3, const float* __restrict__ sumsq,
                             float* __restrict__ mean) {
    float scale = rsqrtf(*sumsq);
    size_t i = (size_t)blockIdx.x * blockDim.x + threadIdx.x;
    size_t stride = (size_t)gridDim.x * blockDim.x;
    for (; i < UD; i += stride) mean[i] += t3[i] * scale;
}

// ---------------- gamma[b] = dot(mean_u[users[b]], mean_i[items[b]]) / 16 ----------------
__global__ void k_gather(const float* __restrict__ mean, const int* __restrict__ users,
                         const int* __restrict__ items, float* __restrict__ out) {
    int wave = (int)((blockIdx.x * blockDim.x + threadIdx.x) >> 5);
    int lane = threadIdx.x & 31;
    if (wave >= BATCH) return;
    int u  = users[wave];
    int it = items[wave];
    const float* mu = mean + (size_t)u * DIM;
    const float* mi = mean + ((size_t)N_USERS + it) * DIM;
    float part = mu[lane] * mi[lane] + mu[lane + 32] * mi[lane + 32];
    for (int off = 16; off > 0; off >>= 1)
        part += __shfl_down(part, off, 32);
    if (lane == 0) out[wave] = part * (1.0f / 16.0f);   // (acc/4)·(acc/4)
}

extern "C" void kernel_launch(void* const* d_in, const int* in_sizes, int n_in,
                              void* d_out, int out_size, void* d_ws, size_t ws_size,
                              hipStream_t stream) {
    const float* user_emb = (const float*)d_in[0];
    const float* item_emb = (const float*)d_in[1];
    const int*   sub_src  = (const int*)  d_in[2];
    const int*   sub_dst  = (const int*)  d_in[3];
    const float* sub_val  = (const float*)d_in[4];
    const int*   soc_src  = (const int*)  d_in[5];
    const int*   soc_dst  = (const int*)  d_in[6];
    const float* soc_val  = (const float*)d_in[7];
    const float* fc1_w    = (const float*)d_in[8];
    const float* fc1_b    = (const float*)d_in[9];
    const float* fc2_w    = (const float*)d_in[10];
    const float* fc2_b    = (const float*)d_in[11];
    const float* fc3_w    = (const float*)d_in[12];
    const float* fc3_b    = (const float*)d_in[13];
    const int*   users    = (const int*)  d_in[14];
    const int*   items    = (const int*)  d_in[15];
    float* out = (float*)d_out;

    // workspace layout (floats)
    float* ws = (float*)d_ws;
    float* persA = ws;                       // 4*ND
    float* persB = persA + 4 * ND;           // 4*ND
    float* usocA = persB + 4 * ND;           // UD
    float* usocB = usocA + UD;               // UD
    float* mean  = usocB + UD;               // ND
    float* cbuf  = mean  + ND;               // N_USERS*192 = 3*UD
    float* t1    = cbuf  + 3 * UD;           // N_USERS*192 = 3*UD
    float* sumsq = t1    + 3 * UD;           // 1
    float* t2    = cbuf;                     // alias: c is dead after fc1
    float* t3    = cbuf + UD;                // alias

    const int TB = 256;
    const int gridND    = (int)((ND + TB - 1) / TB);
    const int gridUD    = (int)((UD + TB - 1) / TB);
    const int spmmBlks  = 2048;              // 16384 waves, grid-stride over edges
    const int mTiles    = N_USERS / 16;      // 6250 waves, one row-strip each
    const int gemmBlks  = (mTiles * 32 + TB - 1) / TB;

    k_init<<<gridND, TB, 0, stream>>>(user_emb, item_emb, persA, mean, usocA);

    for (int layer = 0; layer < 3; ++layer) {
        float* pin  = (layer & 1) ? persB : persA;
        float* pout = (layer & 1) ? persA : persB;
        float* uin  = (layer & 1) ? usocB : usocA;
        float* uout = (layer & 1) ? usocA : usocB;

        k_zero<<<(int)((4 * ND + TB - 1) / TB), TB, 0, stream>>>(pout, 4 * ND);
        k_zero<<<gridUD, TB, 0, stream>>>(uout, UD);

        // persona SpMMs (blockIdx.y = persona)
        k_spmm<<<dim3(spmmBlks, PPERS), TB, 0, stream>>>(
            sub_src, sub_dst, sub_val, pin, pout, E_SUB, E_SUB, ND);
        // social SpMM
        k_spmm<<<dim3(spmmBlks, 1), TB, 0, stream>>>(
            soc_src, soc_dst, soc_val, uin, uout, E_SOC, 0, (size_t)0);

        k_sum_build<<<gridND, TB, 0, stream>>>(pout, uout, mean, cbuf, sumsq);

        // fc1: (100000 x 192) @ (192 x 192), tanh -> two NT=6 column-half launches
        k_gemm_wmma<6><<<gemmBlks, TB, 0, stream>>>(
            cbuf, fc1_w, fc1_b, t1, N_USERS, K3D, K3D, 0,  1, nullptr);
        k_gemm_wmma<6><<<gemmBlks, TB, 0, stream>>>(
            cbuf, fc1_w, fc1_b, t1, N_USERS, K3D, K3D, 96, 1, nullptr);
        // fc2: (100000 x 192) @ (192 x 64), tanh        -> NT = 4
        k_gemm_wmma<4><<<gemmBlks, TB, 0, stream>>>(
            t1, fc2_w, fc2_b, t2, N_USERS, K3D, DIM, 0, 1, nullptr);
        // fc3: (100000 x 64) @ (64 x 64), linear+sumsq  -> NT = 4
        k_gemm_wmma<4><<<gemmBlks, TB, 0, stream>>>(
            t2, fc3_w, fc3_b, t3, N_USERS, DIM, DIM, 0, 0, sumsq);

        k_norm_accum<<<gridUD, TB, 0, stream>>>(t3, sumsq, mean);
    }

    k_gather<<<(BATCH * 32 + TB - 1) / TB, TB, 0, stream>>>(mean, users, items, out);
}